// BlockV2_11038065951390
// MI455X (gfx1250) — compile-verified
//
#include <hip/hip_runtime.h>
#include <math.h>

// Problem constants (from reference)
#define B_   32
#define L_   32768
#define H_   32
#define N2_  8
#define T_   64              // chunk length
#define NC_  (L_ / T_)       // 512 chunks

typedef __attribute__((ext_vector_type(2))) float v2f;
typedef __attribute__((ext_vector_type(8))) float v8f;

// ---------------------------------------------------------------------------
// Setup 1: K[h, 0..63] and P[h, 0..63, 0..15]
//   K[h,l]      = 2*Re( sum_n Ct_n * r_n^l )
//   P[h,l,2n]   = 2*Re( Ct_n * r_n^{l+1} ),  P[h,l,2n+1] = -2*Im( Ct_n * r_n^{l+1} )
// One thread per (h, l): 2048 threads.
// ---------------------------------------------------------------------------
__global__ __launch_bounds__(256) void ssm_setup_kp(
    const float* __restrict__ log_dt, const float* __restrict__ C_re,
    const float* __restrict__ C_im, const float* __restrict__ log_A_real,
    const float* __restrict__ A_imag, float* __restrict__ Kg,
    float* __restrict__ Pg)
{
    int t = blockIdx.x * 256 + threadIdx.x;   // t = h*64 + l
    int l = t & 63;
    int h = t >> 6;
    float dt = expf(log_dt[h]);
    float ksum = 0.f;
    for (int n = 0; n < N2_; ++n) {
        float Are = -expf(log_A_real[h * N2_ + n]);
        float Aim = A_imag[h * N2_ + n];
        float dAre = Are * dt, dAim = Aim * dt;
        // r = exp(dtA)
        float er  = expf(dAre);
        float rre = er * cosf(dAim), rim = er * sinf(dAim);
        // Ct = (C_re + i C_im) * (r - 1) / A
        float nre = rre - 1.f, nim = rim;
        float den = Are * Are + Aim * Aim;
        float qre = (nre * Are + nim * Aim) / den;
        float qim = (nim * Are - nre * Aim) / den;
        float cre = C_re[h * N2_ + n], cim = C_im[h * N2_ + n];
        float Ctre = cre * qre - cim * qim;
        float Ctim = cre * qim + cim * qre;
        // r^l = exp(l*dtA)
        float el  = expf(dAre * (float)l);
        float plr = el * cosf(dAim * (float)l), pli = el * sinf(dAim * (float)l);
        ksum += 2.f * (Ctre * plr - Ctim * pli);
        // r^{l+1}
        float e1  = expf(dAre * (float)(l + 1));
        float p1r = e1 * cosf(dAim * (float)(l + 1));
        float p1i = e1 * sinf(dAim * (float)(l + 1));
        float vre = Ctre * p1r - Ctim * p1i;
        float vim = Ctre * p1i + Ctim * p1r;
        Pg[(h * T_ + l) * 16 + 2 * n]     =  2.f * vre;
        Pg[(h * T_ + l) * 16 + 2 * n + 1] = -2.f * vim;
    }
    Kg[h * T_ + l] = ksum;
}

// ---------------------------------------------------------------------------
// Setup 2: FiLM gamma/beta = cond @ film_W + film_b. One thread per (b,h).
// ---------------------------------------------------------------------------
__global__ __launch_bounds__(256) void ssm_film(
    const float* __restrict__ cond, const float* __restrict__ film_W,
    const float* __restrict__ film_b, float* __restrict__ gamma,
    float* __restrict__ beta)
{
    int t = blockIdx.x * 256 + threadIdx.x;   // t = b*32 + h
    int h = t & 31, b = t >> 5;
    float g  = film_b[h];
    float be = film_b[H_ + h];
    for (int c = 0; c < 4; ++c) {
        float cv = cond[b * 4 + c];
        g  += cv * film_W[c * 2 * H_ + h];
        be += cv * film_W[c * 2 * H_ + H_ + h];
    }
    gamma[b * H_ + h] = g;
    beta [b * H_ + h] = be;
}

// ---------------------------------------------------------------------------
// Phase 1: chunk summaries via Horner:  contrib_n = sum_m r_n^{63-m} u[cT+m]
// One thread per (b, c, h), h fastest for coalesced x reads. 524288 threads.
// Writes Sbuf[b][h][c][2n(+1)] = contrib (re, im).
// ---------------------------------------------------------------------------
__global__ __launch_bounds__(256) void ssm_chunk(
    const float* __restrict__ x, const float* __restrict__ log_dt,
    const float* __restrict__ log_A_real, const float* __restrict__ A_imag,
    float* __restrict__ Sbuf)
{
    int t = blockIdx.x * 256 + threadIdx.x;   // (b, c, h)
    int h = t & 31;
    int c = (t >> 5) & (NC_ - 1);
    int b = t >> 14;
    float dt = expf(log_dt[h]);
    float rre[N2_], rim[N2_], sre[N2_], sim[N2_];
    #pragma unroll
    for (int n = 0; n < N2_; ++n) {
        float dAre = -expf(log_A_real[h * N2_ + n]) * dt;
        float dAim = A_imag[h * N2_ + n] * dt;
        float e = expf(dAre);
        rre[n] = e * cosf(dAim);
        rim[n] = e * sinf(dAim);
        sre[n] = 0.f; sim[n] = 0.f;
    }
    int xbase = (b * L_ + c * T_) * H_ + h;
    for (int m = 0; m < T_; ++m) {
        float u = x[xbase + m * H_];
        #pragma unroll
        for (int n = 0; n < N2_; ++n) {
            float tr = fmaf(rre[n], sre[n], fmaf(-rim[n], sim[n], u));
            float ti = fmaf(rre[n], sim[n], rim[n] * sre[n]);
            sre[n] = tr; sim[n] = ti;
        }
    }
    int sb = ((b * H_ + h) * NC_ + c) * (2 * N2_);
    #pragma unroll
    for (int n = 0; n < N2_; ++n) {
        Sbuf[sb + 2 * n]     = sre[n];
        Sbuf[sb + 2 * n + 1] = sim[n];
    }
}

// ---------------------------------------------------------------------------
// Phase 2: in-place inter-chunk scan. One thread per (b,h,n): 8192 threads.
//   s <- 0;  for c: tmp = Sbuf[c]; Sbuf[c] = s (state ENTERING chunk c);
//            s = r^64 * s + tmp
// ---------------------------------------------------------------------------
__global__ __launch_bounds__(256) void ssm_scan(
    const float* __restrict__ log_dt, const float* __restrict__ log_A_real,
    const float* __restrict__ A_imag, float* __restrict__ Sbuf)
{
    int t = blockIdx.x * 256 + threadIdx.x;   // (b, h, n)
    int n = t & 7;
    int h = (t >> 3) & 31;
    int b = t >> 8;
    float dt = expf(log_dt[h]);
    float dAre = -expf(log_A_real[h * N2_ + n]) * dt;
    float dAim = A_imag[h * N2_ + n] * dt;
    // r^T = exp(T * dtA)
    float e  = expf((float)T_ * dAre);
    float Rr = e * cosf((float)T_ * dAim);
    float Ri = e * sinf((float)T_ * dAim);
    float sre = 0.f, sim = 0.f;
    int base = ((b * H_ + h) * NC_) * (2 * N2_) + 2 * n;
    for (int c = 0; c < NC_; ++c) {
        int idx = base + c * (2 * N2_);
        float cre = Sbuf[idx], cim = Sbuf[idx + 1];
        Sbuf[idx]     = sre;
        Sbuf[idx + 1] = sim;
        float nr = fmaf(Rr, sre, fmaf(-Ri, sim, cre));
        float ni = fmaf(Rr, sim, fmaf(Ri, sre, cim));
        sre = nr; sim = ni;
    }
}

// ---------------------------------------------------------------------------
// Phase 3 (main, WMMA): per (h, chunk) a wave computes the 64x32 tile
//   Y = Kmat_h @ U + P_h @ S + D_h*U, then FiLM + tanh-GELU, store (B,L,H).
// Block = 256 threads = 8 waves, all on the same h (Kmat/P shared in LDS).
// f32 WMMA 16x16x4, K-chained: 16 steps (Toeplitz) + 4 steps (carry state).
// ---------------------------------------------------------------------------
#define KM_STRIDE 66   // LDS padding to dodge bank conflicts
#define P_STRIDE  18

__global__ __launch_bounds__(256) void ssm_main(
    const float* __restrict__ x, const float* __restrict__ Kg,
    const float* __restrict__ Pg, const float* __restrict__ gamma,
    const float* __restrict__ beta, const float* __restrict__ Dg,
    const float* __restrict__ Sbuf, float* __restrict__ out)
{
    __shared__ float Kmat[T_][KM_STRIDE];   // 64x64 lower-tri Toeplitz (padded)
    __shared__ float Pl[T_][P_STRIDE];      // 64x16 carry-projection (padded)

    const int bx = blockIdx.x;              // 0..2047
    const int h = bx >> 6;                  // 64 blocks per h
    const int cbase = (bx & 63) * 8;        // 8 chunks per block
    const int tid = threadIdx.x;

    for (int i = tid; i < T_ * T_; i += 256) {
        int r0 = i >> 6, c0 = i & 63;
        Kmat[r0][c0] = (r0 >= c0) ? Kg[h * T_ + (r0 - c0)] : 0.f;
    }
    for (int i = tid; i < T_ * 16; i += 256)
        Pl[i >> 4][i & 15] = Pg[h * T_ * 16 + i];
    __syncthreads();

    const int wave = tid >> 5, lane = tid & 31;
    const int c  = cbase + wave;
    const int lo = lane & 15;
    const int hi = lane >> 4;               // 0 for lanes 0-15, 1 for 16-31
    const float Dh = Dg[h];

    for (int nt = 0; nt < 2; ++nt) {        // two 16-batch N-tiles
        const int b = lo + nt * 16;
        const float gma = gamma[b * H_ + h];
        const float bta = beta [b * H_ + h];
        const int xbase = (b * L_ + c * T_) * H_ + h;

        // B-fragments of U (16 K-steps of 4): lanes 0-15 hold rows {0,1} of
        // each 4-group, lanes 16-31 rows {2,3} (ISA 7.12.2 striping).
        v2f bu[16];
        #pragma unroll
        for (int k = 0; k < 16; ++k) {
            int row = 4 * k + 2 * hi;
            bu[k].x = x[xbase + row * H_];
            bu[k].y = x[xbase + (row + 1) * H_];
        }
        // B-fragments of S (4 K-steps of 4), consecutive floats -> b64 loads
        const int sbase = ((b * H_ + h) * NC_ + c) * (2 * N2_);
        v2f bs[4];
        #pragma unroll
        for (int k = 0; k < 4; ++k) {
            int j = 4 * k + 2 * hi;
            bs[k].x = Sbuf[sbase + j];
            bs[k].y = Sbuf[sbase + j + 1];
        }

        #pragma unroll
        for (int mt = 0; mt < 4; ++mt) {    // four 16-row M-tiles
            v8f acc = {};
            // carry-state term: P_h (64x16) @ S (16x32), K = 16 -> 4 WMMAs
            #pragma unroll
            for (int k = 0; k < 4; ++k) {
                v2f a;
                a.x = Pl[16 * mt + lo][4 * k + 2 * hi];
                a.y = Pl[16 * mt + lo][4 * k + 2 * hi + 1];
                acc = __builtin_amdgcn_wmma_f32_16x16x4_f32(
                    false, a, false, bs[k], (short)0, acc, false, false);
            }
            // intra-chunk Toeplitz: Kmat (64x64) @ U (64x32), K = 64 -> 16 WMMAs
            #pragma unroll
            for (int k = 0; k < 16; ++k) {
                v2f a;
                a.x = Kmat[16 * mt + lo][4 * k + 2 * hi];
                a.y = Kmat[16 * mt + lo][4 * k + 2 * hi + 1];
                acc = __builtin_amdgcn_wmma_f32_16x16x4_f32(
                    false, a, false, bu[k], (short)0, acc, false, false);
            }
            // Epilogue: skip connection, FiLM, tanh-GELU, store.
            // C/D layout: VGPR r holds M = 16*mt + r (lanes 0-15) or +8 (16-31).
            #pragma unroll
            for (int r = 0; r < 8; ++r) {
                int m = 16 * mt + r + 8 * hi;
                int oidx = (b * L_ + c * T_ + m) * H_ + h;
                float uv = x[oidx];
                float y = acc[r] + Dh * uv;
                y = fmaf(gma, y, bta);
                float inner = 0.7978845608028654f * (y + 0.044715f * y * y * y);
                y = 0.5f * y * (1.f + tanhf(inner));
                out[oidx] = y;
            }
        }
    }
}

// ---------------------------------------------------------------------------
// Launcher. Workspace layout (floats):
//   Kg    @ 0        (H*64            = 2048)
//   Pg    @ 2048     (H*64*16         = 32768)
//   gamma @ 34816    (B*H             = 1024)
//   beta  @ 35840    (B*H             = 1024)
//   Sbuf  @ 36864    (B*H*NC*2*N2     = 8388608)   total ~33.7 MB
// ---------------------------------------------------------------------------
extern "C" void kernel_launch(void* const* d_in, const int* in_sizes, int n_in,
                              void* d_out, int out_size, void* d_ws, size_t ws_size,
                              hipStream_t stream)
{
    const float* x          = (const float*)d_in[0];
    const float* cond       = (const float*)d_in[1];
    const float* log_dt     = (const float*)d_in[2];
    const float* C_re       = (const float*)d_in[3];
    const float* C_im       = (const float*)d_in[4];
    const float* log_A_real = (const float*)d_in[5];
    const float* A_imag     = (const float*)d_in[6];
    const float* Dg         = (const float*)d_in[7];
    const float* film_W     = (const float*)d_in[8];
    const float* film_b     = (const float*)d_in[9];
    float* out = (float*)d_out;

    float* w     = (float*)d_ws;
    float* Kg    = w;
    float* Pg    = w + 2048;
    float* gamma = w + 34816;
    float* beta  = w + 35840;
    float* Sbuf  = w + 36864;

    // Setup (tiny)
    ssm_setup_kp<<<8, 256, 0, stream>>>(log_dt, C_re, C_im, log_A_real, A_imag, Kg, Pg);
    ssm_film<<<4, 256, 0, stream>>>(cond, film_W, film_b, gamma, beta);
    // Phase 1: chunk summaries (B*NC*H threads)
    ssm_chunk<<<(B_ * NC_ * H_) / 256, 256, 0, stream>>>(x, log_dt, log_A_real, A_imag, Sbuf);
    // Phase 2: in-place inter-chunk scan (B*H*N2 threads)
    ssm_scan<<<(B_ * H_ * N2_) / 256, 256, 0, stream>>>(log_dt, log_A_real, A_imag, Sbuf);
    // Phase 3: WMMA intra-chunk matmuls + epilogue (H*NC waves)
    ssm_main<<<(H_ * NC_) / 8, 256, 0, stream>>>(x, Kg, Pg, gamma, beta, Dg, Sbuf, out);
}